// ChamferLoss_49246095016551
// MI455X (gfx1250) — compile-verified
//
#include <hip/hip_runtime.h>

typedef float v2f __attribute__((ext_vector_type(2)));
typedef float v8f __attribute__((ext_vector_type(8)));

#define BATCH 8
#define NPTS  8192
#define TILES (NPTS / 16)                    // 512 column tiles per batch
#define ROWBLKS (NPTS / 32)                  // 256 row blocks (32 rows) per batch
#define WAVES_PER_DIR (BATCH * ROWBLKS)      // 2048 waves per direction

// One wave per 32-row block of P (two 16-row WMMA tiles). Sweeps all 16-col
// tiles of Q. Inner loop tracks min_j(|q_j|^2 - 2*p.q_j) per row via
// V_WMMA_F32_16X16X4_F32 (K=3 zero-padded to 4) + fma + min; |p|^2 is added
// once per row at the end (min/clamp commute with the constant shift).
__global__ __launch_bounds__(256) void chamfer_rowmin_kernel(
    const float* __restrict__ P,   // (B, NPTS, 3) — rows (the "query" set)
    const float* __restrict__ Q,   // (B, NPTS, 3) — cols (the "searched" set)
    float* __restrict__ partial)   // WAVES_PER_DIR floats
{
    const int tid  = blockIdx.x * blockDim.x + threadIdx.x;
    const int lane = threadIdx.x & 31;
    const int wave = tid >> 5;
    const int batch  = wave / ROWBLKS;
    const int rowBlk = wave % ROWBLKS;
    const int row0   = rowBlk * 32;

    const float* Pb = P + (size_t)batch * NPTS * 3;
    const float* Qb = Q + (size_t)batch * NPTS * 3;

    const int hi  = lane >> 4;   // 0: lanes 0-15 carry K=0,1 | 1: lanes 16-31 carry K=2,3
    const int l16 = lane & 15;

    // ---- Two A fragments: rows [row0, row0+16) and [row0+16, row0+32) ----
    v2f afragA, afragB;
    {
        const int ra = row0 + l16;
        const float ax = Pb[ra * 3 + 0], ay = Pb[ra * 3 + 1], az = Pb[ra * 3 + 2];
        afragA.x = hi ? az : ax;      // K=0 (x) or K=2 (z)
        afragA.y = hi ? 0.0f : ay;    // K=1 (y) or K=3 (pad)
        const int rb = row0 + 16 + l16;
        const float bx = Pb[rb * 3 + 0], by = Pb[rb * 3 + 1], bz = Pb[rb * 3 + 2];
        afragB.x = hi ? bz : bx;
        afragB.y = hi ? 0.0f : by;
    }

    // ---- |p|^2 for the 8 rows each accumulator register covers (used only at end) ----
    // C/D layout: VGPR r holds row M = r (lanes 0-15) or M = r+8 (lanes 16-31).
    float a2A[8], a2B[8];
#pragma unroll
    for (int r = 0; r < 8; ++r) {
        const int ma = row0 + r + hi * 8;
        float x = Pb[ma * 3 + 0], y = Pb[ma * 3 + 1], z = Pb[ma * 3 + 2];
        a2A[r] = x * x + y * y + z * z;
        const int mb = ma + 16;
        x = Pb[mb * 3 + 0]; y = Pb[mb * 3 + 1]; z = Pb[mb * 3 + 2];
        a2B[r] = x * x + y * y + z * z;
    }

    float rminA[8], rminB[8];
#pragma unroll
    for (int r = 0; r < 8; ++r) { rminA[r] = __builtin_inff(); rminB[r] = __builtin_inff(); }

    // ---- Sweep all column tiles (no divergence: EXEC all-1s for WMMA) ----
    for (int ct = 0; ct < TILES; ++ct) {
        const int col = ct * 16 + l16;
        const float bx = Qb[col * 3 + 0];
        const float by = Qb[col * 3 + 1];
        const float bz = Qb[col * 3 + 2];

        // B fragment: 4x16 f32 (N striped over lanes%16, K over VGPR + half)
        v2f bfrag;
        bfrag.x = hi ? bz : bx;
        bfrag.y = hi ? 0.0f : by;

        const float b2 = bx * bx + by * by + bz * bz;  // |q|^2 for column N = l16

        v8f czero = {};
        v8f abA = __builtin_amdgcn_wmma_f32_16x16x4_f32(
            false, afragA, false, bfrag, (short)0, czero, false, false);
        v8f abB = __builtin_amdgcn_wmma_f32_16x16x4_f32(
            false, afragB, false, bfrag, (short)0, czero, false, false);

#pragma unroll
        for (int r = 0; r < 8; ++r) {
            // track b2 - 2*ab; the constant |p|^2 shift is applied after the min
            rminA[r] = fminf(rminA[r], __builtin_fmaf(-2.0f, abA[r], b2));
            rminB[r] = fminf(rminB[r], __builtin_fmaf(-2.0f, abB[r], b2));
        }
    }

    // ---- Min across the 16 lanes of each half (columns of the tile) ----
#pragma unroll
    for (int r = 0; r < 8; ++r) {
        float va = rminA[r], vb = rminB[r];
        va = fminf(va, __shfl_xor(va, 8, 16));
        vb = fminf(vb, __shfl_xor(vb, 8, 16));
        va = fminf(va, __shfl_xor(va, 4, 16));
        vb = fminf(vb, __shfl_xor(vb, 4, 16));
        va = fminf(va, __shfl_xor(va, 2, 16));
        vb = fminf(vb, __shfl_xor(vb, 2, 16));
        va = fminf(va, __shfl_xor(va, 1, 16));
        vb = fminf(vb, __shfl_xor(vb, 1, 16));
        rminA[r] = va;   // every lane in the half holds its half's row-min
        rminB[r] = vb;
    }

    // ---- Add |p|^2, clamp at 0, sum this wave's 32 row-mins, emit one partial ----
    float s = 0.0f;
#pragma unroll
    for (int r = 0; r < 8; ++r) {
        s += fmaxf(rminA[r] + a2A[r], 0.0f);
        s += fmaxf(rminB[r] + a2B[r], 0.0f);
    }
    // lane l and l^16 hold the two halves' sums (rows 0-7 and 8-15 of each tile)
    s += __shfl_xor(s, 16, 32);
    if (lane == 0) partial[wave] = s;
}

// Deterministic fixed-order reduction of all partials -> scalar mean.
__global__ __launch_bounds__(256) void chamfer_finalize_kernel(
    const float* __restrict__ partial, float* __restrict__ out)
{
    __shared__ float sm[256];
    float s = 0.0f;
    for (int i = threadIdx.x; i < 2 * WAVES_PER_DIR; i += 256) s += partial[i];
    sm[threadIdx.x] = s;
    __syncthreads();
    for (int off = 128; off > 0; off >>= 1) {
        if (threadIdx.x < off) sm[threadIdx.x] += sm[threadIdx.x + off];
        __syncthreads();
    }
    if (threadIdx.x == 0) out[0] = sm[0] * (1.0f / (float)(BATCH * NPTS));
}

extern "C" void kernel_launch(void* const* d_in, const int* in_sizes, int n_in,
                              void* d_out, int out_size, void* d_ws, size_t ws_size,
                              hipStream_t stream) {
    const float* gts   = (const float*)d_in[0];   // (8, 8192, 3) f32
    const float* preds = (const float*)d_in[1];   // (8, 8192, 3) f32
    float* out     = (float*)d_out;
    float* partial = (float*)d_ws;                // 2*2048 floats = 16 KB scratch

    const int threads = 256;
    const int blocks  = (WAVES_PER_DIR * 32) / threads;  // 256 blocks, 8 waves each

    // direction 1: for each gt, min over preds
    chamfer_rowmin_kernel<<<blocks, threads, 0, stream>>>(gts, preds, partial);
    // direction 2: for each pred, min over gts
    chamfer_rowmin_kernel<<<blocks, threads, 0, stream>>>(preds, gts, partial + WAVES_PER_DIR);
    // combine: (sum of all clamped row-mins) / (B*N), identical divisor both dirs
    chamfer_finalize_kernel<<<1, 256, 0, stream>>>(partial, out);
}